// DoubleGraphConvNet_55052890800551
// MI455X (gfx1250) — compile-verified
//
#include <hip/hip_runtime.h>
#include <math.h>

typedef __attribute__((ext_vector_type(2))) float v2f;
typedef __attribute__((ext_vector_type(8))) float v8f;

#define NG 50000
#define EG 800000
#define NS 10000
#define ES 160000
#define BB 8

// ---------------- zero fill ----------------
__global__ void fill_zero_kernel(float* __restrict__ p, int n) {
  int i = blockIdx.x * blockDim.x + threadIdx.x;
  if (i < n) p[i] = 0.0f;
}

// ---------------- edge scatter-add: agg[dst] += x[src] ----------------
// One thread per (edge, 4 features). c4shift = log2(C/4). float4 vector load
// (global_load_b128) + 4 atomic f32 adds to consecutive addresses.
__global__ void scatter_add_kernel(const float* __restrict__ x,
                                   const int* __restrict__ src,
                                   const int* __restrict__ dst,
                                   float* __restrict__ agg,
                                   int E, int c4shift) {
  long long idx = (long long)blockIdx.x * blockDim.x + threadIdx.x;
  long long total = (long long)E << c4shift;
  if (idx >= total) return;
  const int C = 4 << c4shift;
  const int e  = (int)(idx >> c4shift);
  const int f4 = (int)(idx & ((1 << c4shift) - 1)) << 2;
  const int s = src[e];
  const int d = dst[e];
  const float4 v = *(const float4*)(x + (size_t)s * C + f4);
  float* ap = agg + (size_t)d * C + f4;
  atomicAdd(ap + 0, v.x);
  atomicAdd(ap + 1, v.y);
  atomicAdd(ap + 2, v.z);
  atomicAdd(ap + 3, v.w);
}

// ---------------- fused GraphConv GEMM + bias + ELU (f32 WMMA) -------------
// out[n,o] = sum_c agg[n,c]*Wr[o,c] + sum_c x[n,c]*Wn[o,c] + b[o]; optional ELU.
// One wave computes a 16x64 output strip: 4 column tiles share each A fragment
// (1 A load amortized over 4 V_WMMA_F32_16X16X4_F32).
// A (16x4 f32): lane m=lane&15 is row; VGPR0=K{0|2}, VGPR1=K{1|3} per lane half
//   -> float2 at k0 + 2*(lane>>4).
// B (4x16): lane n=lane&15 selects output channel row of W; same K split.
// C/D (16x16): VGPR v -> row v + 8*(lane>>4), col lane&15.
__global__ __launch_bounds__(32)
void graphconv_wmma_kernel(const float* __restrict__ agg,
                           const float* __restrict__ xin,
                           const float* __restrict__ Wr,
                           const float* __restrict__ Wn,
                           const float* __restrict__ bias,
                           float* __restrict__ out,
                           int Cin, int Cout, int applyElu) {
  const int lane = threadIdx.x;            // 0..31, all active (EXEC all-1s)
  const int m    = lane & 15;
  const int kb   = (lane >> 4) << 1;       // 0 or 2
  const int row  = blockIdx.x * 16 + m;    // node row (A)
  const int col0 = blockIdx.y * 64;        // first of 4 output-channel tiles

  const float* arow = agg + (size_t)row * Cin + kb;
  const float* xrow = xin + (size_t)row * Cin + kb;

  const float* wr0 = Wr + (size_t)(col0      + m) * Cin + kb;
  const float* wr1 = Wr + (size_t)(col0 + 16 + m) * Cin + kb;
  const float* wr2 = Wr + (size_t)(col0 + 32 + m) * Cin + kb;
  const float* wr3 = Wr + (size_t)(col0 + 48 + m) * Cin + kb;
  const float* wn0 = Wn + (size_t)(col0      + m) * Cin + kb;
  const float* wn1 = Wn + (size_t)(col0 + 16 + m) * Cin + kb;
  const float* wn2 = Wn + (size_t)(col0 + 32 + m) * Cin + kb;
  const float* wn3 = Wn + (size_t)(col0 + 48 + m) * Cin + kb;

  v8f acc0 = {}, acc1 = {}, acc2 = {}, acc3 = {};

#pragma unroll 4
  for (int k0 = 0; k0 < Cin; k0 += 4) {
    v2f a  = *(const v2f*)(arow + k0);
    v2f b0 = *(const v2f*)(wr0 + k0);
    v2f b1 = *(const v2f*)(wr1 + k0);
    v2f b2 = *(const v2f*)(wr2 + k0);
    v2f b3 = *(const v2f*)(wr3 + k0);
    acc0 = __builtin_amdgcn_wmma_f32_16x16x4_f32(false, a, false, b0, (short)0, acc0, false, false);
    acc1 = __builtin_amdgcn_wmma_f32_16x16x4_f32(false, a, false, b1, (short)0, acc1, false, false);
    acc2 = __builtin_amdgcn_wmma_f32_16x16x4_f32(false, a, false, b2, (short)0, acc2, false, false);
    acc3 = __builtin_amdgcn_wmma_f32_16x16x4_f32(false, a, false, b3, (short)0, acc3, false, false);
  }
#pragma unroll 4
  for (int k0 = 0; k0 < Cin; k0 += 4) {
    v2f a  = *(const v2f*)(xrow + k0);
    v2f b0 = *(const v2f*)(wn0 + k0);
    v2f b1 = *(const v2f*)(wn1 + k0);
    v2f b2 = *(const v2f*)(wn2 + k0);
    v2f b3 = *(const v2f*)(wn3 + k0);
    acc0 = __builtin_amdgcn_wmma_f32_16x16x4_f32(false, a, false, b0, (short)0, acc0, false, false);
    acc1 = __builtin_amdgcn_wmma_f32_16x16x4_f32(false, a, false, b1, (short)0, acc1, false, false);
    acc2 = __builtin_amdgcn_wmma_f32_16x16x4_f32(false, a, false, b2, (short)0, acc2, false, false);
    acc3 = __builtin_amdgcn_wmma_f32_16x16x4_f32(false, a, false, b3, (short)0, acc3, false, false);
  }

  const int rbase = blockIdx.x * 16 + ((lane >> 4) << 3);
  v8f accs[4] = {acc0, acc1, acc2, acc3};
#pragma unroll
  for (int t = 0; t < 4; ++t) {
    const int outcol = col0 + t * 16 + (lane & 15);
    const float bval = bias[outcol];
#pragma unroll
    for (int v = 0; v < 8; ++v) {
      float val = accs[t][v] + bval;
      if (applyElu) val = (val > 0.0f) ? val : expm1f(val);
      out[(size_t)(rbase + v) * Cout + outcol] = val;
    }
  }
}

// ---------------- batch mean-pool (C = 192, float4 per thread) ----------------
__global__ void pool_accum_kernel(const float* __restrict__ x,
                                  const int* __restrict__ batch,
                                  float* __restrict__ sum, int N, int C) {
  const int C4 = C >> 2;
  long long idx = (long long)blockIdx.x * blockDim.x + threadIdx.x;
  if (idx >= (long long)N * C4) return;
  int n  = (int)(idx / C4);
  int f4 = (int)(idx - (long long)n * C4) << 2;
  const float4 v = *(const float4*)(x + (size_t)n * C + f4);
  float* sp = sum + batch[n] * C + f4;
  atomicAdd(sp + 0, v.x);
  atomicAdd(sp + 1, v.y);
  atomicAdd(sp + 2, v.z);
  atomicAdd(sp + 3, v.w);
}

__global__ void pool_count_kernel(const int* __restrict__ batch,
                                  float* __restrict__ cnt, int N) {
  int i = blockIdx.x * blockDim.x + threadIdx.x;
  if (i < N) atomicAdd(&cnt[batch[i]], 1.0f);
}

// ---------------- concat [mean1(192) | mean2(192) | point(64)] ----------------
__global__ void concat_kernel(const float* __restrict__ s1, const float* __restrict__ c1,
                              const float* __restrict__ s2, const float* __restrict__ c2,
                              const float* __restrict__ point, float* __restrict__ feat) {
  int idx = blockIdx.x * blockDim.x + threadIdx.x;
  if (idx >= BB * 448) return;
  int b = idx / 448, j = idx - b * 448;
  float v;
  if (j < 192)       v = s1[b * 192 + j]        / fmaxf(c1[b], 1.0f);
  else if (j < 384)  v = s2[b * 192 + (j-192)]  / fmaxf(c2[b], 1.0f);
  else               v = point[b * 64 + (j-384)];
  feat[idx] = v;
}

// ---------------- small dense layer: out = in @ W^T + b (opt relu) ----------------
__global__ void linear_kernel(const float* __restrict__ in, const float* __restrict__ W,
                              const float* __restrict__ b, float* __restrict__ out,
                              int Bn, int K, int O, int relu) {
  int idx = blockIdx.x * blockDim.x + threadIdx.x;
  if (idx >= Bn * O) return;
  int bi = idx / O, o = idx - bi * O;
  const float* xr = in + (size_t)bi * K;
  const float* wr = W  + (size_t)o  * K;
  float s = b[o];
  for (int k = 0; k < K; ++k) s += xr[k] * wr[k];
  if (relu) s = fmaxf(s, 0.0f);
  out[idx] = s;
}

extern "C" void kernel_launch(void* const* d_in, const int* in_sizes, int n_in,
                              void* d_out, int out_size, void* d_ws, size_t ws_size,
                              hipStream_t stream) {
  (void)in_sizes; (void)n_in; (void)out_size; (void)ws_size;

  const float* graph_x    = (const float*)d_in[0];
  const float* subgraph_x = (const float*)d_in[1];
  const float* point      = (const float*)d_in[2];
  const int*   g_ei       = (const int*)d_in[3];   // [2, EG] row-major: src, dst
  const int*   g_batch    = (const int*)d_in[4];
  const int*   s_ei       = (const int*)d_in[5];   // [2, ES]
  const int*   s_batch    = (const int*)d_in[6];
  // d_in[7..15]  = gW1r,gW1n,gB1, gW2r,gW2n,gB2, gW3r,gW3n,gB3
  // d_in[16..24] = sW1r,... same layout
  const float* l1W = (const float*)d_in[25];
  const float* l1b = (const float*)d_in[26];
  const float* l2W = (const float*)d_in[27];
  const float* l2b = (const float*)d_in[28];
  const float* l3W = (const float*)d_in[29];
  const float* l3b = (const float*)d_in[30];

  // Workspace layout (floats)
  float* ws   = (float*)d_ws;
  float* agg  = ws;                    // NG*256
  float* bufX = ws + 12800000;         // NG*256
  float* bufY = ws + 25600000;         // NG*256
  float* smll = ws + 38400000;
  float* sum1 = smll;                  // 8*192
  float* cnt1 = smll + 1536;           // 8
  float* sum2 = smll + 1544;           // 8*192
  float* cnt2 = smll + 3080;           // 8
  float* feat = smll + 3088;           // 8*448
  float* m1   = smll + 6672;           // 8*600
  float* m2   = smll + 11472;          // 8*256

  const int Cins[3]  = {64, 128, 256};
  const int Couts[3] = {128, 256, 192};
  const int c4sh[3]  = {4, 5, 6};      // log2(Cin/4)

  for (int br = 0; br < 2; ++br) {
    const int N = (br == 0) ? NG : NS;
    const int E = (br == 0) ? EG : ES;
    const float* x0    = (br == 0) ? graph_x : subgraph_x;
    const int*   ei    = (br == 0) ? g_ei : s_ei;
    const int*   batch = (br == 0) ? g_batch : s_batch;
    float* sum = (br == 0) ? sum1 : sum2;
    const int wbase = 7 + br * 9;

    const float* cur = x0;
    float* outs[3] = {bufX, bufY, bufX};   // layer3 reuses bufX (free by then)
    for (int li = 0; li < 3; ++li) {
      const float* Wr = (const float*)d_in[wbase + li * 3 + 0];
      const float* Wn = (const float*)d_in[wbase + li * 3 + 1];
      const float* bb = (const float*)d_in[wbase + li * 3 + 2];
      const int Cin = Cins[li], Cout = Couts[li];

      const int nAgg = N * Cin;
      fill_zero_kernel<<<(nAgg + 255) / 256, 256, 0, stream>>>(agg, nAgg);

      const long long tot = (long long)E << c4sh[li];
      scatter_add_kernel<<<(int)((tot + 255) / 256), 256, 0, stream>>>(
          cur, ei, ei + E, agg, E, c4sh[li]);

      graphconv_wmma_kernel<<<dim3(N / 16, Cout / 64), 32, 0, stream>>>(
          agg, cur, Wr, Wn, bb, outs[li], Cin, Cout, 1);
      cur = outs[li];
    }

    // mean pool over batch (sum + count, contiguous 1544-float region zeroed)
    fill_zero_kernel<<<7, 256, 0, stream>>>(sum, 1544);
    const int nPool = N * (192 / 4);
    pool_accum_kernel<<<(nPool + 255) / 256, 256, 0, stream>>>(cur, batch, sum, N, 192);
    pool_count_kernel<<<(N + 255) / 256, 256, 0, stream>>>(batch, sum + 1536, N);
  }

  concat_kernel<<<(BB * 448 + 255) / 256, 256, 0, stream>>>(sum1, cnt1, sum2, cnt2, point, feat);

  linear_kernel<<<(BB * 600 + 255) / 256, 256, 0, stream>>>(feat, l1W, l1b, m1, BB, 448, 600, 1);
  linear_kernel<<<(BB * 256 + 255) / 256, 256, 0, stream>>>(m1,   l2W, l2b, m2, BB, 600, 256, 1);
  linear_kernel<<<(BB * 64  + 255) / 256, 256, 0, stream>>>(m2,   l3W, l3b, (float*)d_out, BB, 256, 64, 0);
}